// TripletLoss_63728724738653
// MI455X (gfx1250) — compile-verified
//
#include <hip/hip_runtime.h>
#include <math.h>

// ---------------------------------------------------------------------------
// TripletLoss, batch-hard mining, fused.
//   dot = A·Negᵀ via V_WMMA_F32_16X16X4_F32 (fp32-faithful to the reference),
//   staged through double-buffered LDS with GLOBAL_LOAD_ASYNC_TO_LDS_B128
//   (ASYNCcnt), row-min epilogue -> atomic uint-min (never materializes 8192²).
// ---------------------------------------------------------------------------

typedef float v2f __attribute__((ext_vector_type(2)));
typedef float v4f __attribute__((ext_vector_type(4)));
typedef float v8f __attribute__((ext_vector_type(8)));

#define TL_MARGIN 0.25f
#define TL_EPS    1e-16f
#define TL_D      256    // feature dim (fixed by the reference)

#define BM   64          // rows per workgroup (2 wave-rows of 32)
#define BN   128         // cols per workgroup (4 wave-cols of 32)
#define KC   32          // K chunk staged in LDS
#define LDA  36          // padded LDS row stride (floats): rows -> disjoint 4-bank groups
#define ASZ  (BM * LDA)  // floats per A buffer (2304)
#define NSZ  (BN * LDA)  // floats per N buffer (4608)

// --- CDNA5 async copy: global -> LDS, 16B per lane, tracked by ASYNCcnt -----
__device__ __forceinline__ void async_ld16(unsigned lds_addr, const float* g) {
  asm volatile("global_load_async_to_lds_b128 %0, %1, off"
               :: "v"(lds_addr), "v"((unsigned long long)(uintptr_t)g)
               : "memory");
}
__device__ __forceinline__ void wait_async0() {
  asm volatile("s_wait_asynccnt 0x0" ::: "memory");
}

__device__ __forceinline__ v8f wmma4(v2f a, v2f b, v8f c) {
  return __builtin_amdgcn_wmma_f32_16x16x4_f32(false, a, false, b, (short)0, c,
                                               false, false);
}

// ---------------------------------------------------------------------------
// Kernel 1: per-row stats, one wave32 per row.
// ---------------------------------------------------------------------------
__global__ __launch_bounds__(256)
void tl_stats_kernel(const float* __restrict__ A, const float* __restrict__ P,
                     const float* __restrict__ Ng,
                     float* __restrict__ anorm, float* __restrict__ nnorm,
                     float* __restrict__ dpos, unsigned int* __restrict__ minsq,
                     int B) {
  const int row  = (blockIdx.x * blockDim.x + threadIdx.x) >> 5;
  const int lane = threadIdx.x & 31;
  if (row >= B) return;

  const v4f* a4 = reinterpret_cast<const v4f*>(A  + row * TL_D) + lane * 2;
  const v4f* p4 = reinterpret_cast<const v4f*>(P  + row * TL_D) + lane * 2;
  const v4f* n4 = reinterpret_cast<const v4f*>(Ng + row * TL_D) + lane * 2;

  float sa = 0.f, sd = 0.f, sn = 0.f;
#pragma unroll
  for (int j = 0; j < 2; ++j) {
    v4f av = a4[j], pv = p4[j], nv = n4[j];
#pragma unroll
    for (int k = 0; k < 4; ++k) {
      float a = av[k], p = pv[k], n = nv[k];
      sa += a * a;
      float d = a - p;
      sd += d * d;
      sn += n * n;
    }
  }
#pragma unroll
  for (int m = 16; m >= 1; m >>= 1) {
    sa += __shfl_xor(sa, m, 32);
    sd += __shfl_xor(sd, m, 32);
    sn += __shfl_xor(sn, m, 32);
  }
  if (lane == 0) {
    anorm[row] = sa;
    nnorm[row] = sn;
    dpos[row]  = sqrtf(fmaxf(sd, TL_EPS));
    minsq[row] = 0x7f800000u;  // +inf bits; all candidates are >= EPS > 0
  }
}

// ---------------------------------------------------------------------------
// Epilogue helper for one 16-row group covered by two column tiles (c0,c1):
// sq = anorm + nnorm - 2*dot, clamp EPS, min across tj and across the 16 lanes
// of each half, then atomic uint-min into minsq[row].
// C layout: c[i] on lane l holds tile row (i + 8*(l>>4)), tile col (l&15).
// ---------------------------------------------------------------------------
__device__ __forceinline__ void tile_epilogue(const v8f& c0, const v8f& c1,
                                              const float* __restrict__ anorm,
                                              float nn0, float nn1, int rowBase,
                                              int half, int l15,
                                              unsigned int* __restrict__ minsq) {
  float sq[8];
#pragma unroll
  for (int i = 0; i < 8; ++i) {
    const float an = anorm[rowBase + 8 * half + i];
    const float s  = fminf(an + nn0 - 2.0f * c0[i], an + nn1 - 2.0f * c1[i]);
    sq[i] = fmaxf(s, TL_EPS);
  }
#pragma unroll
  for (int i = 0; i < 8; ++i) {
#pragma unroll
    for (int m = 8; m >= 1; m >>= 1)  // xor 1,2,4,8 stays within each 16-lane half
      sq[i] = fminf(sq[i], __shfl_xor(sq[i], m, 32));
  }
  if (l15 == 0) {  // lane 0 -> rows +0..7, lane 16 -> rows +8..15
#pragma unroll
    for (int i = 0; i < 8; ++i)
      atomicMin(&minsq[rowBase + 8 * half + i], __float_as_uint(sq[i]));
  }
}

// ---------------------------------------------------------------------------
// Kernel 2: fused GEMM + row-min.
//   256 threads = 8 waves arranged 2(M) x 4(N); each wave computes a 32x32
//   region as 2x2 WMMA tiles (4 x v8f accumulators) so each pair of LDS
//   fragment loads feeds 2 WMMAs.  K staged in double-buffered LDS via
//   async-to-LDS copies overlapped with compute.
// ---------------------------------------------------------------------------
__global__ __launch_bounds__(256)
void tl_minsq_kernel(const float* __restrict__ A, const float* __restrict__ Ng,
                     const float* __restrict__ anorm,
                     const float* __restrict__ nnorm,
                     unsigned int* __restrict__ minsq) {
  __shared__ float lds[2 * ASZ + 2 * NSZ];  // 55,296 B

  const int tid   = threadIdx.x;
  const int lane  = tid & 31;
  const int wave  = tid >> 5;     // 0..7
  const int half  = lane >> 4;    // 0: K=(0,1), 1: K=(2,3)   (ISA 16x4 fp32 layout)
  const int l15   = lane & 15;
  const int waveM = wave >> 2;    // 0..1  (32 rows each)
  const int waveN = wave & 3;     // 0..3  (32 cols each)
  const int rBlock = blockIdx.x * BM;
  const int cBlock = blockIdx.y * BN;

  const unsigned ldsBase = (unsigned)(uintptr_t)&lds[0];  // low 32 bits == DS offset

  // Stage one KC-wide chunk of A (64xKC) and N (128xKC) into buffer `buf`.
  auto stage = [&](int buf, int k0) {
    const unsigned aB = ldsBase + (unsigned)(buf * ASZ * 4);
    const unsigned nB = ldsBase + (unsigned)((2 * ASZ + buf * NSZ) * 4);
#pragma unroll
    for (int it = 0; it < 2; ++it) {           // A: 64 rows x 8 float4
      const int idx = it * 256 + tid;
      const int r = idx >> 3, c = (idx & 7) << 2;
      async_ld16(aB + (unsigned)((r * LDA + c) * 4),
                 A + (rBlock + r) * TL_D + k0 + c);
    }
#pragma unroll
    for (int it = 0; it < 4; ++it) {           // N: 128 rows x 8 float4
      const int idx = it * 256 + tid;
      const int r = idx >> 3, c = (idx & 7) << 2;
      async_ld16(nB + (unsigned)((r * LDA + c) * 4),
                 Ng + (cBlock + r) * TL_D + k0 + c);
    }
  };

  v8f acc00 = {}, acc01 = {}, acc10 = {}, acc11 = {};

  stage(0, 0);
  int cur = 0;
  for (int k0 = 0; k0 < TL_D; k0 += KC) {
    wait_async0();        // my async copies into lds[cur] have landed
    __syncthreads();      // everyone's copies landed; prior chunk fully consumed
    if (k0 + KC < TL_D) stage(cur ^ 1, k0 + KC);  // prefetch next chunk

    const float* Ab  = &lds[cur * ASZ];
    const float* Nb  = &lds[2 * ASZ + cur * NSZ];
    const float* a0p = Ab + (waveM * 32 + l15) * LDA;
    const float* a1p = a0p + 16 * LDA;
    const float* b0p = Nb + (waveN * 32 + l15) * LDA;
    const float* b1p = b0p + 16 * LDA;

#pragma unroll
    for (int kk = 0; kk < KC; kk += 4) {
      const int ks = kk + (half << 1);
      v2f a0 = *reinterpret_cast<const v2f*>(a0p + ks);  // conflict-free ds_load_b64
      v2f a1 = *reinterpret_cast<const v2f*>(a1p + ks);
      v2f b0 = *reinterpret_cast<const v2f*>(b0p + ks);
      v2f b1 = *reinterpret_cast<const v2f*>(b1p + ks);
      acc00 = wmma4(a0, b0, acc00);
      acc01 = wmma4(a0, b1, acc01);
      acc10 = wmma4(a1, b0, acc10);
      acc11 = wmma4(a1, b1, acc11);
    }
    cur ^= 1;
  }

  // ---- epilogue: 2 row groups x (2 col tiles combined) ----
  const float nn0 = nnorm[cBlock + waveN * 32 + l15];
  const float nn1 = nnorm[cBlock + waveN * 32 + 16 + l15];
  const int rowBase = rBlock + waveM * 32;
  tile_epilogue(acc00, acc01, anorm, nn0, nn1, rowBase,      half, l15, minsq);
  tile_epilogue(acc10, acc11, anorm, nn0, nn1, rowBase + 16, half, l15, minsq);
}

// ---------------------------------------------------------------------------
// Kernel 3: losses + mean-over-active (fallback to plain mean).
// ---------------------------------------------------------------------------
__global__ __launch_bounds__(256)
void tl_finalize_kernel(const float* __restrict__ dpos,
                        const unsigned int* __restrict__ minsq,
                        float* __restrict__ out, int B) {
  __shared__ float sSum[256];
  __shared__ float sCnt[256];
  const int tid = threadIdx.x;
  float sum = 0.f, cnt = 0.f;
  for (int b = tid; b < B; b += 256) {
    const float hardest = sqrtf(__uint_as_float(minsq[b]));
    const float loss = dpos[b] - hardest + TL_MARGIN;
    if (loss > 0.f) { sum += loss; cnt += 1.f; }
  }
  sSum[tid] = sum;
  sCnt[tid] = cnt;
  __syncthreads();
  for (int s = 128; s > 0; s >>= 1) {
    if (tid < s) { sSum[tid] += sSum[tid + s]; sCnt[tid] += sCnt[tid + s]; }
    __syncthreads();
  }
  if (tid == 0) {
    const float c = sCnt[0], s = sSum[0];
    out[0] = (c > 0.f) ? (s / c) : (s / (float)B);
  }
}

// ---------------------------------------------------------------------------
extern "C" void kernel_launch(void* const* d_in, const int* in_sizes, int n_in,
                              void* d_out, int out_size, void* d_ws, size_t ws_size,
                              hipStream_t stream) {
  (void)n_in; (void)out_size; (void)ws_size;
  const float* A  = (const float*)d_in[0];
  const float* P  = (const float*)d_in[1];
  const float* Ng = (const float*)d_in[2];
  const int B = in_sizes[0] / TL_D;   // 8192

  // workspace: [minsq u32 x B][anorm f32 x B][nnorm f32 x B][dpos f32 x B]
  unsigned int* minsq = (unsigned int*)d_ws;
  float* anorm = (float*)d_ws + B;
  float* nnorm = (float*)d_ws + 2 * B;
  float* dpos  = (float*)d_ws + 3 * B;

  const int g1 = (B * 32 + 255) / 256;  // one wave per row
  tl_stats_kernel<<<g1, 256, 0, stream>>>(A, P, Ng, anorm, nnorm, dpos, minsq, B);

  dim3 g2(B / BM, B / BN);  // (128, 64) blocks of 256 threads
  tl_minsq_kernel<<<g2, 256, 0, stream>>>(A, Ng, anorm, nnorm, minsq);

  tl_finalize_kernel<<<1, 256, 0, stream>>>(dpos, minsq, (float*)d_out, B);
}